// MultiHeadFactorizedRandomAttention_13280038879453
// MI455X (gfx1250) — compile-verified
//
#include <hip/hip_runtime.h>
#include <hip/hip_bf16.h>

// Problem constants (fixed by the reference).
#define BATCH   4
#define HEADS   16
#define SEQ     1024
#define RANK    64
#define DMODEL  1024
#define HDIM    (DMODEL / HEADS)   // 64
#define BN      (BATCH * SEQ)      // 4096 rows in the GEMMs
#define BHN     (BATCH * HEADS * SEQ)

typedef __attribute__((ext_vector_type(2))) float v2f;
typedef __attribute__((ext_vector_type(8))) float v8f;
typedef int v4i __attribute__((ext_vector_type(4)));
// Pointer types matching the async-to-LDS builtin signature:
//   param0: addrspace(1) int4*  (diagnostic showed "int __device__ *", vector_size 16)
//   param1: addrspace(3) int4*  (v*1 v*3 pattern)
typedef __attribute__((address_space(1))) v4i* gbl_v4i_ptr;
typedef __attribute__((address_space(3))) v4i* lds_v4i_ptr;

// CDNA5 async global->LDS path (ASYNCcnt-tracked, no VGPR round-trip).
#if defined(__has_builtin)
#  if __has_builtin(__builtin_amdgcn_global_load_async_to_lds_b128)
#    define USE_ASYNC_LDS 1
#  endif
#endif
#ifndef USE_ASYNC_LDS
#  define USE_ASYNC_LDS 0
#endif

// ---------------------------------------------------------------------------
// Kernel 1: closed-form softmax coefficients.
// scores row i = diag d_i at col i, 0 elsewhere =>
//   out_i = alpha_i * v_i + beta_i * S,  alpha=(e^d-1)/(e^d+N-1), beta=1/(e^d+N-1)
// ---------------------------------------------------------------------------
__global__ __launch_bounds__(256) void coef_kernel(const float* __restrict__ fl,
                                                   const float* __restrict__ fr,
                                                   float* __restrict__ alpha,
                                                   float* __restrict__ beta) {
    const int idx = blockIdx.x * 256 + threadIdx.x;        // (b,h,n) flattened
    const size_t base = (size_t)idx * RANK;
    float d = 0.f;
#pragma unroll
    for (int r = 0; r < RANK; r += 4) {
        const float4 a = *(const float4*)(fl + base + r);
        const float4 b = *(const float4*)(fr + base + r);
        d += a.x * b.x + a.y * b.y + a.z * b.z + a.w * b.w;
    }
    const float e = expf(d);
    const float inv = 1.0f / (e + (float)(SEQ - 1));
    alpha[idx] = (e - 1.0f) * inv;
    beta[idx]  = inv;
}

// ---------------------------------------------------------------------------
// Kernel 2/5: C[M,N] = A[M,K] @ B[N,K]^T  (both operands K-contiguous, fp32)
// via V_WMMA_F32_16X16X4_F32. Block tile 128x128, K-tile 32, double-buffered
// LDS staging; global->LDS via async-to-LDS when the builtin exists.
// 8 waves (wave32) in 4x2; each wave computes 2x4 16x16 tiles.
// ---------------------------------------------------------------------------
#define TILE_M 128
#define TILE_N 128
#define TILE_K 32
#define LDS_STRIDE 36   // 144B row pitch: 16B-aligned rows, conflict-free banks

__global__ __launch_bounds__(256) void gemm_nt_wmma(const float* __restrict__ A,
                                                    const float* __restrict__ Bm,
                                                    float* __restrict__ C,
                                                    int M, int Nn, int K) {
    __shared__ float As[2][TILE_M * LDS_STRIDE];   // 2 x 18 KB
    __shared__ float Bs[2][TILE_M * LDS_STRIDE];   // 2 x 18 KB  (72 KB total)

    const int tid   = threadIdx.x;
    const int lane  = tid & 31;
    const int wave  = tid >> 5;       // 0..7
    const int waveM = wave >> 1;      // 0..3 -> 32-row band
    const int waveN = wave & 1;       // 0..1 -> 64-col band
    const int m16   = lane & 15;      // M (A) / N (B) index within fragment
    const int g     = lane >> 4;      // K-pair group: k = 2g, 2g+1

    const int rowBase = blockIdx.y * TILE_M;
    const int colBase = blockIdx.x * TILE_N;

#if !USE_ASYNC_LDS
    float4 regA[4], regB[4];
#endif

    // Issue the global->LDS traffic for one 128x32 tile pair.
    auto stage = [&](int buf, int k0) {
#pragma unroll
        for (int i = 0; i < 4; ++i) {
            const int linear = tid + i * 256;      // 0..1023 float4 slots
            const int r  = linear >> 3;            // row 0..127 (8 float4/row)
            const int cv = (linear & 7) << 2;      // col 0,4,...,28
            const float* gA = A  + (size_t)(rowBase + r) * K + k0 + cv;
            const float* gB = Bm + (size_t)(colBase + r) * K + k0 + cv;
#if USE_ASYNC_LDS
            __builtin_amdgcn_global_load_async_to_lds_b128(
                (gbl_v4i_ptr)gA, (lds_v4i_ptr)&As[buf][r * LDS_STRIDE + cv], 0, 0);
            __builtin_amdgcn_global_load_async_to_lds_b128(
                (gbl_v4i_ptr)gB, (lds_v4i_ptr)&Bs[buf][r * LDS_STRIDE + cv], 0, 0);
#else
            regA[i] = *(const float4*)gA;
            regB[i] = *(const float4*)gB;
#endif
        }
    };

    // Make the staged tile resident in LDS (own wave's part).
    auto commit = [&](int buf) {
#if USE_ASYNC_LDS
        (void)buf;
#  if __has_builtin(__builtin_amdgcn_s_wait_asynccnt)
        __builtin_amdgcn_s_wait_asynccnt(0);
#  else
        asm volatile("s_wait_asynccnt 0" ::: "memory");
#  endif
#else
#pragma unroll
        for (int i = 0; i < 4; ++i) {
            const int linear = tid + i * 256;
            const int r  = linear >> 3;
            const int cv = (linear & 7) << 2;
            *(float4*)(&As[buf][r * LDS_STRIDE + cv]) = regA[i];
            *(float4*)(&Bs[buf][r * LDS_STRIDE + cv]) = regB[i];
        }
#endif
    };

    v8f acc[2][4];
#pragma unroll
    for (int mt = 0; mt < 2; ++mt)
#pragma unroll
        for (int nt = 0; nt < 4; ++nt)
            acc[mt][nt] = (v8f){0.f, 0.f, 0.f, 0.f, 0.f, 0.f, 0.f, 0.f};

    // Prologue: tile 0 into buffer 0.
    stage(0, 0);
    commit(0);
    __syncthreads();

    const int nstages = K / TILE_K;
    for (int s = 0; s < nstages; ++s) {
        const int cur = s & 1;
        const int nxt = cur ^ 1;
        if (s + 1 < nstages) stage(nxt, (s + 1) * TILE_K);   // overlap with WMMAs

        const float* sA = As[cur];
        const float* sB = Bs[cur];
#pragma unroll
        for (int kk = 0; kk < TILE_K; kk += 4) {
            // A fragment: lane holds A[m16, kk+2g .. kk+2g+1]  (ISA 16x4 layout)
            v2f aF[2];
#pragma unroll
            for (int mt = 0; mt < 2; ++mt)
                aF[mt] = *(const v2f*)(&sA[(waveM * 32 + mt * 16 + m16) * LDS_STRIDE + kk + 2 * g]);
            // B fragment: lane holds B[kk+2g .. kk+2g+1, n=m16]
            v2f bF[4];
#pragma unroll
            for (int nt = 0; nt < 4; ++nt)
                bF[nt] = *(const v2f*)(&sB[(waveN * 64 + nt * 16 + m16) * LDS_STRIDE + kk + 2 * g]);
#pragma unroll
            for (int mt = 0; mt < 2; ++mt)
#pragma unroll
                for (int nt = 0; nt < 4; ++nt)
                    acc[mt][nt] = __builtin_amdgcn_wmma_f32_16x16x4_f32(
                        false, aF[mt], false, bF[nt], (short)0, acc[mt][nt], false, false);
        }

        if (s + 1 < nstages) commit(nxt);
        __syncthreads();   // cur fully consumed; nxt fully resident
    }

    // D layout: VGPR v -> M = v + 8g, N = m16.
#pragma unroll
    for (int mt = 0; mt < 2; ++mt)
#pragma unroll
        for (int nt = 0; nt < 4; ++nt)
#pragma unroll
            for (int v = 0; v < 8; ++v) {
                const int r = rowBase + waveM * 32 + mt * 16 + g * 8 + v;
                const int c = colBase + waveN * 64 + nt * 16 + m16;
                C[(size_t)r * Nn + c] = acc[mt][nt][v];
            }
}

// ---------------------------------------------------------------------------
// Kernel 3: per-(b,h) head sums  S[b,h,d] = sum_n value[b,n,h*64+d]
// ---------------------------------------------------------------------------
__global__ __launch_bounds__(256) void headsum_kernel(const float* __restrict__ value,
                                                      float* __restrict__ S) {
    __shared__ float red[256];
    const int bh = blockIdx.x;           // 0..63
    const int b  = bh / HEADS;
    const int h  = bh % HEADS;
    const int t  = threadIdx.x;
    const int d  = t & (HDIM - 1);
    const int c  = t >> 6;               // chunk 0..3
    float acc = 0.f;
    const size_t colBase = (size_t)h * HDIM + d;
    for (int n = c * 256; n < (c + 1) * 256; ++n)
        acc += value[((size_t)b * SEQ + n) * DMODEL + colBase];
    red[t] = acc;
    __syncthreads();
    if (c == 0)
        S[(size_t)bh * HDIM + d] = red[d] + red[64 + d] + red[128 + d] + red[192 + d];
}

// ---------------------------------------------------------------------------
// Kernel 4: in-place mix  value[b,n,f] = alpha[b,h,n]*value + beta[b,h,n]*S[b,h,d]
// ---------------------------------------------------------------------------
__global__ __launch_bounds__(256) void mix_kernel(float* __restrict__ value,
                                                  const float* __restrict__ alpha,
                                                  const float* __restrict__ beta,
                                                  const float* __restrict__ S) {
    const size_t idx = (size_t)blockIdx.x * 256 + threadIdx.x;   // over B*N*DMODEL
    const int f = (int)(idx & (DMODEL - 1));
    const int n = (int)((idx >> 10) & (SEQ - 1));
    const int b = (int)(idx >> 20);
    const int h = f >> 6;
    const int d = f & (HDIM - 1);
    const int bh = b * HEADS + h;
    const size_t ci = (size_t)bh * SEQ + n;
    value[idx] = alpha[ci] * value[idx] + beta[ci] * S[(size_t)bh * HDIM + d];
}

// ---------------------------------------------------------------------------
extern "C" void kernel_launch(void* const* d_in, const int* in_sizes, int n_in,
                              void* d_out, int out_size, void* d_ws, size_t ws_size,
                              hipStream_t stream) {
    (void)in_sizes; (void)n_in; (void)out_size; (void)ws_size;
    const float* x  = (const float*)d_in[0];
    const float* fl = (const float*)d_in[1];
    const float* fr = (const float*)d_in[2];
    const float* Wv = (const float*)d_in[3];
    const float* Wo = (const float*)d_in[4];
    float* out = (float*)d_out;

    float* ws    = (float*)d_ws;
    float* value = ws;                                   // BN * DMODEL
    float* alpha = value + (size_t)BN * DMODEL;          // BHN
    float* beta  = alpha + BHN;                          // BHN
    float* S     = beta + BHN;                           // B*H*HDIM

    // 1) softmax coefficients from the factor diagonal
    coef_kernel<<<BHN / 256, 256, 0, stream>>>(fl, fr, alpha, beta);

    // 2) value = x @ Wv^T   (M=4096, N=1024, K=1024)
    dim3 gemmGrid(DMODEL / TILE_N, BN / TILE_M);
    gemm_nt_wmma<<<gemmGrid, 256, 0, stream>>>(x, Wv, value, BN, DMODEL, DMODEL);

    // 3) per-head sums over the sequence
    headsum_kernel<<<BATCH * HEADS, 256, 0, stream>>>(value, S);

    // 4) in-place attention mix
    mix_kernel<<<(BN * DMODEL) / 256, 256, 0, stream>>>(value, alpha, beta, S);

    // 5) out = mid @ Wo^T
    gemm_nt_wmma<<<gemmGrid, 256, 0, stream>>>(value, Wo, out, BN, DMODEL, DMODEL);
}